// GraphNodes_70179765616701
// MI455X (gfx1250) — compile-verified
//
#include <hip/hip_runtime.h>
#include <hip/hip_bf16.h>

typedef __attribute__((ext_vector_type(2))) float v2f;
typedef __attribute__((ext_vector_type(8))) float v8f;

#define GN 4096
#define GF 256
#define LEAKY 0.2f
#define THRESH 1e-12f

enum { GEMM_NN = 0, GEMM_NT = 1, GEMM_TN = 2 };
enum { EPI_NONE = 0, EPI_ROWSCALE = 1, EPI_BIAS_LEAKY = 2 };

// ---- fragment loaders ------------------------------------------------------
// A fragment 16x4 f32: lane (m16, half) holds A[m, ka], A[m, ka+1], ka = k+2*half
template <int TRANS>
__device__ __forceinline__ v2f load_afrag(const float* __restrict__ A, int ka,
                                          int m, int lda) {
  v2f a;
  if (TRANS == GEMM_TN) {
    a.x = A[(size_t)ka * lda + m];
    a.y = A[(size_t)(ka + 1) * lda + m];
  } else {
    const float* p = &A[(size_t)m * lda + ka];
    a.x = p[0];
    a.y = p[1];  // contiguous -> global_load_b64
  }
  return a;
}

// B fragment 4x16 f32: lane (n16, half) holds B[ka, n], B[ka+1, n]
template <int TRANS>
__device__ __forceinline__ v2f load_bfrag(const float* __restrict__ B, int ka,
                                          int n, int ldb) {
  v2f b;
  if (TRANS == GEMM_NT) {
    const float* p = &B[(size_t)n * ldb + ka];
    b.x = p[0];
    b.y = p[1];  // contiguous -> global_load_b64
  } else {
    b.x = B[(size_t)ka * ldb + n];
    b.y = B[(size_t)(ka + 1) * ldb + n];
  }
  return b;
}

// ---------------------------------------------------------------------------
// fp32 WMMA GEMM with distance-2 software pipeline.
// Wave tile 16x64 (1 A-frag x 4 B-frags x v_wmma_f32_16x16x4_f32),
// block = 8 waves = 128x64.  Fragments are double-buffered so WMMAs for step
// k only wait on loads issued two steps back while step k+4's loads fly.
// ---------------------------------------------------------------------------
template <int TRANS, int EPI>
__global__ __launch_bounds__(256) void wmma_gemm(
    const float* __restrict__ A, const float* __restrict__ B,
    float* __restrict__ C, const float* __restrict__ epi_vec,
    int K, int lda, int ldb, int ldc) {
  const int lane = threadIdx.x & 31;
  const int wave = threadIdx.x >> 5;
  const int m16  = lane & 15;
  const int half = lane >> 4;

  const int m0 = blockIdx.y * 128 + wave * 16;
  const int n0 = blockIdx.x * 64;
  const int mr = m0 + m16;
  const int k2 = 2 * half;

  v8f acc[4] = {};

  // prologue: fill both pipeline stages (k = 0 and k = 4); K >= 8 always here
  v2f a0 = load_afrag<TRANS>(A, 0 + k2, mr, lda);
  v2f b0[4];
#pragma unroll
  for (int w = 0; w < 4; ++w)
    b0[w] = load_bfrag<TRANS>(B, 0 + k2, n0 + 16 * w + m16, ldb);

  v2f a1 = load_afrag<TRANS>(A, 4 + k2, mr, lda);
  v2f b1[4];
#pragma unroll
  for (int w = 0; w < 4; ++w)
    b1[w] = load_bfrag<TRANS>(B, 4 + k2, n0 + 16 * w + m16, ldb);

  // main loop, hand-unrolled x2 so each stage's buffer index is static
  for (int k = 0; k < K; k += 8) {
    // ---- stage 0: compute k, prefetch k+8 into buffer 0 ----
#pragma unroll
    for (int w = 0; w < 4; ++w)
      acc[w] = __builtin_amdgcn_wmma_f32_16x16x4_f32(
          false, a0, false, b0[w], (short)0, acc[w], false, false);
    {
      const int kn = (k + 8 < K) ? (k + 8) : 0;  // wrap: in-bounds dummy
      a0 = load_afrag<TRANS>(A, kn + k2, mr, lda);
#pragma unroll
      for (int w = 0; w < 4; ++w)
        b0[w] = load_bfrag<TRANS>(B, kn + k2, n0 + 16 * w + m16, ldb);
    }

    // ---- stage 1: compute k+4, prefetch k+12 into buffer 1 ----
#pragma unroll
    for (int w = 0; w < 4; ++w)
      acc[w] = __builtin_amdgcn_wmma_f32_16x16x4_f32(
          false, a1, false, b1[w], (short)0, acc[w], false, false);
    {
      const int kn = (k + 12 < K) ? (k + 12) : 0;
      a1 = load_afrag<TRANS>(A, kn + k2, mr, lda);
#pragma unroll
      for (int w = 0; w < 4; ++w)
        b1[w] = load_bfrag<TRANS>(B, kn + k2, n0 + 16 * w + m16, ldb);
    }
  }

  // C/D layout: VGPR r <-> row (r + 8*half), col = lane%16
#pragma unroll
  for (int w = 0; w < 4; ++w) {
    const int nc = n0 + 16 * w + m16;
#pragma unroll
    for (int r = 0; r < 8; ++r) {
      const int row = m0 + r + 8 * half;
      float v = acc[w][r];
      if (EPI == EPI_ROWSCALE) v *= epi_vec[row];
      if (EPI == EPI_BIAS_LEAKY) {
        v += epi_vec[nc];
        v = (v >= 0.f) ? v : LEAKY * v;
      }
      C[(size_t)row * ldc + nc] = v;
    }
  }
}

// ---------------------------------------------------------------------------
// sq[i] = sum_k x[i,k]^2   (one wave32 per row)
// ---------------------------------------------------------------------------
__global__ __launch_bounds__(256) void row_sumsq(const float* __restrict__ x,
                                                 float* __restrict__ sq) {
  const int row  = blockIdx.x * 8 + (threadIdx.x >> 5);
  const int lane = threadIdx.x & 31;
  float s = 0.f;
  for (int k = lane; k < GF; k += 32) {
    const float v = x[(size_t)row * GF + k];
    s += v * v;
  }
#pragma unroll
  for (int off = 16; off > 0; off >>= 1) s += __shfl_xor(s, off, 32);
  if (lane == 0) sq[row] = s;
}

// ---------------------------------------------------------------------------
// In-place per-row: G[i,j] -> softmax_j( exp(-sqrt(max(sq_i+sq_j-2G, eps))) )
// ---------------------------------------------------------------------------
__global__ __launch_bounds__(256) void softmax_rows(float* __restrict__ G,
                                                    const float* __restrict__ sq) {
  const int row = blockIdx.x;
  const int tid = threadIdx.x;
  const float sqi = sq[row];
  float* grow = G + (size_t)row * GN;

  float v[16];
  float m = -3.0e38f;
#pragma unroll
  for (int c = 0; c < 16; ++c) {
    const int j = tid + c * 256;
    float d2 = sqi + sq[j] - 2.f * grow[j];
    d2 = fmaxf(d2, THRESH);
    const float val = __expf(-__builtin_sqrtf(d2));
    v[c] = val;
    m = fmaxf(m, val);
  }

  __shared__ float red[256];
  red[tid] = m;
  __syncthreads();
#pragma unroll
  for (int s = 128; s > 0; s >>= 1) {
    if (tid < s) red[tid] = fmaxf(red[tid], red[tid + s]);
    __syncthreads();
  }
  m = red[0];
  __syncthreads();

  float sum = 0.f;
#pragma unroll
  for (int c = 0; c < 16; ++c) {
    v[c] = __expf(v[c] - m);
    sum += v[c];
  }
  red[tid] = sum;
  __syncthreads();
#pragma unroll
  for (int s = 128; s > 0; s >>= 1) {
    if (tid < s) red[tid] += red[tid + s];
    __syncthreads();
  }
  const float inv = 1.f / red[0];
#pragma unroll
  for (int c = 0; c < 16; ++c) grow[tid + c * 256] = v[c] * inv;
}

// ---------------------------------------------------------------------------
// Deterministic two-pass column sum of A (bit-stable across graph replays).
// ---------------------------------------------------------------------------
__global__ __launch_bounds__(256) void colsum_partial(const float* __restrict__ Am,
                                                      float* __restrict__ part) {
  const int col = blockIdx.x * 256 + threadIdx.x;
  const int rb  = blockIdx.y;  // 32 row chunks of 128
  float s = 0.f;
  const int r0 = rb * 128;
  for (int i = 0; i < 128; ++i) s += Am[(size_t)(r0 + i) * GN + col];
  part[(size_t)rb * GN + col] = s;
}

__global__ __launch_bounds__(256) void colsum_finish(const float* __restrict__ part,
                                                     float* __restrict__ inv) {
  const int col = blockIdx.x * 256 + threadIdx.x;
  float s = 0.f;
#pragma unroll
  for (int rb = 0; rb < 32; ++rb) s += part[(size_t)rb * GN + col];
  inv[col] = 1.f / s;
}

// ---------------------------------------------------------------------------
extern "C" void kernel_launch(void* const* d_in, const int* in_sizes, int n_in,
                              void* d_out, int out_size, void* d_ws, size_t ws_size,
                              hipStream_t stream) {
  const float* x    = (const float*)d_in[0];  // [4096, 256]
  const float* W    = (const float*)d_in[1];  // [256, 256]
  const float* bias = (const float*)d_in[2];  // [256]
  float* out = (float*)d_out;                 // [4096, 256]

  char* ws = (char*)d_ws;
  float* G    = (float*)ws;                          // N*N (64 MB), becomes A
  float* s    = (float*)(ws + (size_t)GN * GN * 4);  // N*F (4 MB)
  float* u    = s + (size_t)GN * GF;                 // N*F (4 MB)
  float* sq   = u + (size_t)GN * GF;                 // N
  float* ainv = sq + GN;                             // N
  float* part = ainv + GN;                           // 32*N (512 KB)

  // 1) row squared norms
  row_sumsq<<<GN / 8, 256, 0, stream>>>(x, sq);

  // 2) G = x * x^T  (M=N=4096, K=256)  -- WMMA NT
  wmma_gemm<GEMM_NT, EPI_NONE><<<dim3(GN / 64, GN / 128), 256, 0, stream>>>(
      x, x, G, nullptr, GF, GF, GF, GN);

  // 3) s = x * W  (M=4096, N=256, K=256)  -- WMMA NN
  wmma_gemm<GEMM_NN, EPI_NONE><<<dim3(GF / 64, GN / 128), 256, 0, stream>>>(
      x, W, s, nullptr, GF, GF, GF, GF);

  // 4) row softmax of exp(-dist): G -> A in place
  softmax_rows<<<GN, 256, 0, stream>>>(G, sq);

  // 5) 1 / column sums of A (deterministic two-pass)
  colsum_partial<<<dim3(GN / 256, 32), 256, 0, stream>>>(G, part);
  colsum_finish<<<GN / 256, 256, 0, stream>>>(part, ainv);

  // 6) u = diag(1/Alsum) * (A^T * s)  (K=4096) -- WMMA TN + rowscale
  wmma_gemm<GEMM_TN, EPI_ROWSCALE><<<dim3(GF / 64, GN / 128), 256, 0, stream>>>(
      G, s, u, ainv, GN, GN, GF, GF);

  // 7) out = leakyrelu(A * u + bias)  (K=4096) -- WMMA NN + epilogue
  wmma_gemm<GEMM_NN, EPI_BIAS_LEAKY><<<dim3(GF / 64, GN / 128), 256, 0, stream>>>(
      G, u, out, bias, GN, GN, GF, GF);
}